// TopkLayer2d_83434034692101
// MI455X (gfx1250) — compile-verified
//
#include <hip/hip_runtime.h>

#define HX   512
#define WX   512
#define NEUR 32
#define S2   16
#define NHH  509
#define NWW  509
#define NZ   (NHH * NWW)
#define EPSF 1e-8f
#define NEGINF (-3.0e38f)

typedef float v2f __attribute__((ext_vector_type(2)));
typedef float v8f __attribute__((ext_vector_type(8)));

// 1/(sqrt(x)+eps) via native v_sqrt_f32 + v_rcp_f32 (no IEEE div expansion)
__device__ __forceinline__ float inv_norm(float ss) {
    return __builtin_amdgcn_rcpf(__builtin_amdgcn_sqrtf(ss) + EPSF);
}

__global__ __launch_bounds__(256) void
topk_layer2d_wmma(const float* __restrict__ x,
                  const float* __restrict__ Wz,
                  float* __restrict__ out)
{
    __shared__ float xch[8 * 32];   // per-wave 32-float transpose slot

    const int lane = threadIdx.x & 31;
    const int wid  = threadIdx.x >> 5;
    const int z    = blockIdx.x * 8 + wid;     // one wave per patch z
    const bool active = (z < NZ);              // wave-uniform

    const int t   = lane >> 4;                 // K-half select (A/B layout)
    const int col = lane & 15;                 // A row / D column

    v8f acc0 = {};   // neurons 0..15
    v8f acc1 = {};   // neurons 16..31

    if (active) {
        // ---- patch: lane d<16 holds p[d] = x[pr + d/4, pc + d%4] ----
        const int pr = z / NWW;
        const int pc = z % NWW;
        float pv = 0.f;
        if (lane < 16)
            pv = x[(pr + (lane >> 2)) * WX + (pc + (lane & 3))];
        float ps = pv * pv;
        #pragma unroll
        for (int off = 16; off; off >>= 1)
            ps += __shfl_xor(ps, off, 32);
        const float pvn = pv * inv_norm(ps);   // normalized patch value (lanes 0..15)

        // ---- load A (each Wz element touched exactly once, nontemporal) ----
        const float* wbase = Wz + (size_t)z * (NEUR * S2);
        v2f a0[4], a1[4];
        #pragma unroll
        for (int c = 0; c < 4; ++c) {
            const float* p0 = wbase + (col)      * 16 + c * 4 + t * 2;
            const float* p1 = wbase + (16 + col) * 16 + c * 4 + t * 2;
            a0[c] = __builtin_nontemporal_load((const v2f*)p0);
            a1[c] = __builtin_nontemporal_load((const v2f*)p1);
        }

        // ---- row norms: my 8 elems + partner(lane^16) = full 16-elem row ----
        float ss0 = 0.f, ss1 = 0.f;
        #pragma unroll
        for (int c = 0; c < 4; ++c) {
            ss0 += a0[c][0]*a0[c][0] + a0[c][1]*a0[c][1];
            ss1 += a1[c][0]*a1[c][0] + a1[c][1]*a1[c][1];
        }
        ss0 += __shfl_xor(ss0, 16, 32);
        ss1 += __shfl_xor(ss1, 16, 32);
        const float r0 = inv_norm(ss0);
        const float r1 = inv_norm(ss1);
        #pragma unroll
        for (int c = 0; c < 4; ++c) { a0[c] *= r0; a1[c] *= r1; }

        // ---- 8x V_WMMA_F32_16X16X4_F32: D columns replicate the matvec ----
        #pragma unroll
        for (int c = 0; c < 4; ++c) {
            const int k0 = c * 4 + t * 2;
            v2f b;
            b[0] = __shfl(pvn, k0,     32);
            b[1] = __shfl(pvn, k0 + 1, 32);
            acc0 = __builtin_amdgcn_wmma_f32_16x16x4_f32(
                       false, a0[c], false, b, (short)0, acc0, false, false);
            acc1 = __builtin_amdgcn_wmma_f32_16x16x4_f32(
                       false, a1[c], false, b, (short)0, acc1, false, false);
        }

        // ---- transpose via LDS: column-0 lanes (0 and 16) own unique rows ----
        if (col == 0) {
            const int base = wid * 32 + t * 8;
            #pragma unroll
            for (int r = 0; r < 8; ++r) {
                xch[base + r]      = acc0[r];   // neurons  t*8 + r
                xch[base + 16 + r] = acc1[r];   // neurons 16 + t*8 + r
            }
        }
    }

    __syncthreads();

    if (active) {
        const float pre = xch[wid * 32 + lane];   // lane l == neuron l

        // ---- top-4 threshold with multiplicity (matches lax.top_k) ----
        float cur = pre;
        float thr = NEGINF;
        #pragma unroll
        for (int k = 0; k < 4; ++k) {
            float m = cur;
            #pragma unroll
            for (int off = 16; off; off >>= 1)
                m = fmaxf(m, __shfl_xor(m, off, 32));
            thr = m;
            unsigned mask = __builtin_amdgcn_ballot_w32(cur == m);
            int low = __builtin_ctz(mask);        // remove exactly one instance
            if (lane == low) cur = NEGINF;
        }

        out[(size_t)z * NEUR + lane] = (pre >= thr) ? pre : 0.0f;
    }
}

extern "C" void kernel_launch(void* const* d_in, const int* in_sizes, int n_in,
                              void* d_out, int out_size, void* d_ws, size_t ws_size,
                              hipStream_t stream)
{
    const float* x  = (const float*)d_in[0];
    const float* Wz = (const float*)d_in[1];
    float* out      = (float*)d_out;

    const int waves_per_block = 8;             // 256 threads, wave32
    const int blocks = (NZ + waves_per_block - 1) / waves_per_block;
    topk_layer2d_wmma<<<blocks, 256, 0, stream>>>(x, Wz, out);
}